// Decoder_22935125360765
// MI455X (gfx1250) — compile-verified
//
#include <hip/hip_runtime.h>
#include <math.h>

typedef float v2f __attribute__((ext_vector_type(2)));
typedef float v8f __attribute__((ext_vector_type(8)));

// ---------------------------------------------------------------------------
// kNN (k=3) over 3D points with LDS-staged source tiles.
// One thread per "up" point; maintains sorted top-3 smallest squared distances.
// ---------------------------------------------------------------------------
template<int TILE>
__global__ void knn3_kernel(const float* __restrict__ pos_up, const float* __restrict__ pos,
                            const int* __restrict__ batch_up, const int* __restrict__ batch,
                            int n_up, int n_src,
                            int* __restrict__ idx_out, float* __restrict__ w_out)
{
    __shared__ float sx[TILE], sy[TILE], sz[TILE];
    __shared__ int   sb[TILE];
    const int tid = threadIdx.x;
    const int n = blockIdx.x * TILE + tid;
    const bool valid = (n < n_up);
    float px = 0.f, py = 0.f, pz = 0.f; int bu = 0;
    if (valid) {
        px = pos_up[3 * n + 0]; py = pos_up[3 * n + 1]; pz = pos_up[3 * n + 2];
        bu = batch_up[n];
    }
    float d0 = INFINITY, d1 = INFINITY, d2 = INFINITY;
    int   i0 = 0, i1 = 0, i2 = 0;
    for (int s = 0; s < n_src; s += TILE) {
        const int j = s + tid;
        if (j < n_src) {
            sx[tid] = pos[3 * j + 0];
            sy[tid] = pos[3 * j + 1];
            sz[tid] = pos[3 * j + 2];
            sb[tid] = batch[j];
        }
        __syncthreads();
        const int lim = min(TILE, n_src - s);
        for (int t = 0; t < lim; ++t) {
            const float dx = px - sx[t], dy = py - sy[t], dz = pz - sz[t];
            float dist = dx * dx + dy * dy + dz * dz;
            if (sb[t] != bu) dist = INFINITY;
            const int j2 = s + t;
            if (dist < d2) {                       // strict '<' keeps earliest index on ties
                if (dist < d1) {
                    d2 = d1; i2 = i1;
                    if (dist < d0) { d1 = d0; i1 = i0; d0 = dist; i0 = j2; }
                    else           { d1 = dist; i1 = j2; }
                } else { d2 = dist; i2 = j2; }
            }
        }
        __syncthreads();
    }
    if (valid) {
        const float w0 = 1.0f / fmaxf(d0, 1e-16f);
        const float w1 = 1.0f / fmaxf(d1, 1e-16f);
        const float w2 = 1.0f / fmaxf(d2, 1e-16f);
        const float inv = 1.0f / (w0 + w1 + w2);
        idx_out[3 * n + 0] = i0; idx_out[3 * n + 1] = i1; idx_out[3 * n + 2] = i2;
        float4 wv; wv.x = w0; wv.y = w1; wv.z = w2; wv.w = inv;
        reinterpret_cast<float4*>(w_out)[n] = wv;
    }
}

// ---------------------------------------------------------------------------
// Gather + inverse-distance interpolate + concat:  hcat[n] = [x_up[n], interp(x_src)]
// One block per up-node; threads stride channels (coalesced row accesses).
// ---------------------------------------------------------------------------
__global__ void interp_concat_kernel(const float* __restrict__ x_up_lvl,  // [n_up, c2]
                                     const float* __restrict__ x_src,     // [n_src, c1]
                                     const int* __restrict__ idx, const float* __restrict__ wbuf,
                                     float* __restrict__ hcat, int n_up, int c1, int c2)
{
    const int n = blockIdx.x;
    const int i0 = idx[3 * n + 0], i1 = idx[3 * n + 1], i2 = idx[3 * n + 2];
    const float4 wv = reinterpret_cast<const float4*>(wbuf)[n];
    const int cin = c1 + c2;
    for (int c = threadIdx.x; c < cin; c += blockDim.x) {
        float v;
        if (c < c2) {
            v = x_up_lvl[(size_t)n * c2 + c];
        } else {
            const int cc = c - c2;
            v = (wv.x * x_src[(size_t)i0 * c1 + cc] +
                 wv.y * x_src[(size_t)i1 * c1 + cc] +
                 wv.z * x_src[(size_t)i2 * c1 + cc]) * wv.w;
        }
        hcat[(size_t)n * cin + c] = v;
    }
}

// ---------------------------------------------------------------------------
// fp32 WMMA GEMM with 2x2 register tiling:
//   Out[M, Ncols] = A[M, K] @ W[Ncols, K]^T + bias
// One wave per 32x32 output tile (4 x 16x16 accumulators). Per K-step of 4:
// 2 A fragments + 2 B fragments (one b64 load each per lane), 4 WMMAs
// -> 8 B/lane per V_WMMA_F32_16X16X4_F32, 2x less load traffic than 1x1.
// A 16x4 layout: lanes 0-15 -> K=0(v0),K=1(v1); lanes 16-31 -> K=2,K=3.
// D layout: vgpr r, lanes 0-15 -> M=r; lanes 16-31 -> M=r+8.
// M multiple of 32, Ncols multiple of 32, K multiple of 4; EXEC all ones.
// ---------------------------------------------------------------------------
__global__ void gemm_wmma_bias(const float* __restrict__ A, const float* __restrict__ W,
                               const float* __restrict__ bias, float* __restrict__ Out,
                               int K, int Ncols)
{
    const int lane = threadIdx.x;
    const int lo = lane & 15;
    const int hi = lane >> 4;          // 0 or 1
    const int m0 = blockIdx.x * 32;
    const int n0 = blockIdx.y * 32;

    const float* arow0 = A + (size_t)(m0 + lo) * K + 2 * hi;
    const float* arow1 = arow0 + (size_t)16 * K;
    const float* wrow0 = W + (size_t)(n0 + lo) * K + 2 * hi;
    const float* wrow1 = wrow0 + (size_t)16 * K;

    v8f acc00 = {}, acc01 = {}, acc10 = {}, acc11 = {};
    for (int k = 0; k < K; k += 4) {
        const v2f a0 = *reinterpret_cast<const v2f*>(arow0 + k);   // 8B aligned
        const v2f a1 = *reinterpret_cast<const v2f*>(arow1 + k);
        const v2f b0 = *reinterpret_cast<const v2f*>(wrow0 + k);
        const v2f b1 = *reinterpret_cast<const v2f*>(wrow1 + k);
        acc00 = __builtin_amdgcn_wmma_f32_16x16x4_f32(false, a0, false, b0, (short)0, acc00, false, false);
        acc01 = __builtin_amdgcn_wmma_f32_16x16x4_f32(false, a0, false, b1, (short)0, acc01, false, false);
        acc10 = __builtin_amdgcn_wmma_f32_16x16x4_f32(false, a1, false, b0, (short)0, acc10, false, false);
        acc11 = __builtin_amdgcn_wmma_f32_16x16x4_f32(false, a1, false, b1, (short)0, acc11, false, false);
    }

    const float bc0 = bias[n0 + lo];
    const float bc1 = bias[n0 + 16 + lo];
    const int r0 = m0 + hi * 8;        // row base for M-tile 0
    const int r1 = m0 + 16 + hi * 8;   // row base for M-tile 1
#pragma unroll
    for (int r = 0; r < 8; ++r) {
        Out[(size_t)(r0 + r) * Ncols + n0 + lo]      = acc00[r] + bc0;
        Out[(size_t)(r0 + r) * Ncols + n0 + 16 + lo] = acc01[r] + bc1;
        Out[(size_t)(r1 + r) * Ncols + n0 + lo]      = acc10[r] + bc0;
        Out[(size_t)(r1 + r) * Ncols + n0 + 16 + lo] = acc11[r] + bc1;
    }
}

// ---------------------------------------------------------------------------
// Per-column sum & sum-of-squares (C is a power of two, C <= 256).
// LDS partial accumulation, then one global atomic per column per block.
// ---------------------------------------------------------------------------
__global__ void colstats_kernel(const float* __restrict__ H, float* __restrict__ stats,
                                int total, int cmask, int per_block)
{
    __shared__ float ssum[256], ssq[256];
    const int C = cmask + 1;
    for (int t = threadIdx.x; t < C; t += blockDim.x) { ssum[t] = 0.f; ssq[t] = 0.f; }
    __syncthreads();
    size_t start = (size_t)blockIdx.x * (size_t)per_block;
    size_t end = start + (size_t)per_block;
    if (end > (size_t)total) end = (size_t)total;
    for (size_t i = start + threadIdx.x; i < end; i += blockDim.x) {
        const float v = H[i];
        const int c = (int)(i & (size_t)cmask);
        atomicAdd(&ssum[c], v);
        atomicAdd(&ssq[c], v * v);
    }
    __syncthreads();
    for (int t = threadIdx.x; t < C; t += blockDim.x) {
        atomicAdd(&stats[t], ssum[t]);
        atomicAdd(&stats[C + t], ssq[t]);
    }
}

__global__ void zero_kernel(float* __restrict__ p, int n)
{
    const int i = blockIdx.x * blockDim.x + threadIdx.x;
    if (i < n) p[i] = 0.f;
}

// ---------------------------------------------------------------------------
// BatchNorm(batch stats) + ReLU, optionally fused residual add.
// stats = [sum(C), sumsq(C)]; var = E[x^2] - E[x]^2 (population, matches jnp.var)
// ---------------------------------------------------------------------------
__global__ void bn_relu_kernel(const float* __restrict__ H, const float* __restrict__ stats,
                               const float* __restrict__ g, const float* __restrict__ be,
                               const float* __restrict__ resid, float* __restrict__ Out,
                               int total, int cmask, float invN)
{
    const int i = blockIdx.x * blockDim.x + threadIdx.x;
    if (i >= total) return;
    const int c = i & cmask;
    const int C = cmask + 1;
    const float mean = stats[c] * invN;
    const float var = stats[C + c] * invN - mean * mean;
    float v = (H[i] - mean) * rsqrtf(var + 1e-5f) * g[c] + be[c];
    v = fmaxf(v, 0.f);
    if (resid) v += resid[i];
    Out[i] = v;
}

// ---------------------------------------------------------------------------
// Host-side orchestration
// d_in order (setup_inputs dict order):
//   [pos0,x0,batch0, pos1,x1,batch1, pos2,x2,batch2, pos3,x3,batch3,
//    then per level i: W1_i, W2_i, b1_i, be1_i, b2_i, be2_i, g1_i, g2_i]
// ---------------------------------------------------------------------------
extern "C" void kernel_launch(void* const* d_in, const int* in_sizes, int n_in,
                              void* d_out, int out_size, void* d_ws, size_t ws_size,
                              hipStream_t stream)
{
    (void)in_sizes; (void)n_in; (void)out_size; (void)ws_size;
    static const int NS_[4] = {512, 2048, 8192, 32768};
    static const int CH[4]  = {512, 256, 128, 64};

    const float* pos[4]; const float* x[4]; const int* bat[4];
    for (int i = 0; i < 4; ++i) {
        pos[i] = (const float*)d_in[3 * i + 0];
        x[i]   = (const float*)d_in[3 * i + 1];
        bat[i] = (const int*)  d_in[3 * i + 2];
    }

    float* ws = (float*)d_ws;
    size_t o = 0;
    float* P0    = ws + o; o += (size_t)2048 * 256;   // level-0 output features
    float* P1    = ws + o; o += (size_t)8192 * 128;   // level-1 output features
    float* hcat  = ws + o; o += (size_t)32768 * 192;  // max concat buffer
    float* h1    = ws + o; o += (size_t)32768 * 64;   // max post-MLP1
    float* h2    = ws + o; o += (size_t)32768 * 64;   // max post-MLP2
    float* wbuf  = ws + o; o += (size_t)32768 * 4;    // knn weights (float4/node)
    float* stats = ws + o; o += 512;                  // [sum(C), sumsq(C)]
    int*   idxbuf = (int*)(ws + o);                   // 3 ints per node

    const float* xcur = x[0];
    float* xnext[3] = {P0, P1, (float*)d_out};

    for (int lvl = 0; lvl < 3; ++lvl) {
        const int n_src = NS_[lvl], n_up = NS_[lvl + 1];
        const int c1 = CH[lvl], c2 = CH[lvl + 1];
        const int cin = c1 + c2, cout = c2;
        const int pb = 12 + 8 * lvl;
        const float* W1  = (const float*)d_in[pb + 0];
        const float* W2  = (const float*)d_in[pb + 1];
        const float* b1  = (const float*)d_in[pb + 2];
        const float* be1 = (const float*)d_in[pb + 3];
        const float* b2  = (const float*)d_in[pb + 4];
        const float* be2 = (const float*)d_in[pb + 5];
        const float* g1  = (const float*)d_in[pb + 6];
        const float* g2  = (const float*)d_in[pb + 7];

        // 1) kNN top-3 + weights
        knn3_kernel<128><<<(n_up + 127) / 128, 128, 0, stream>>>(
            pos[lvl + 1], pos[lvl], bat[lvl + 1], bat[lvl], n_up, n_src, idxbuf, wbuf);

        // 2) interpolate + concat -> hcat [n_up, cin]
        interp_concat_kernel<<<n_up, 256, 0, stream>>>(
            x[lvl + 1], xcur, idxbuf, wbuf, hcat, n_up, c1, c2);

        const int total = n_up * cout;
        const int per_block = 256 * 64;
        const int sblocks = (total + per_block - 1) / per_block;
        const float invN = 1.0f / (float)n_up;

        // 3) GEMM1 (WMMA fp32, 32x32 tiles) + bias -> h1
        zero_kernel<<<2, 256, 0, stream>>>(stats, 2 * cout);
        gemm_wmma_bias<<<dim3(n_up / 32, cout / 32), 32, 0, stream>>>(
            hcat, W1, b1, h1, cin, cout);
        colstats_kernel<<<sblocks, 256, 0, stream>>>(h1, stats, total, cout - 1, per_block);
        bn_relu_kernel<<<(total + 255) / 256, 256, 0, stream>>>(
            h1, stats, g1, be1, nullptr, h1, total, cout - 1, invN);

        // 4) GEMM2 (WMMA fp32, 32x32 tiles) + bias -> h2 ; out = h1 + relu(bn(h2))
        zero_kernel<<<2, 256, 0, stream>>>(stats, 2 * cout);
        gemm_wmma_bias<<<dim3(n_up / 32, cout / 32), 32, 0, stream>>>(
            h1, W2, b2, h2, cout, cout);
        colstats_kernel<<<sblocks, 256, 0, stream>>>(h2, stats, total, cout - 1, per_block);
        bn_relu_kernel<<<(total + 255) / 256, 256, 0, stream>>>(
            h2, stats, g2, be2, h1, xnext[lvl], total, cout - 1, invN);

        xcur = xnext[lvl];
    }
}